// encoding3_54760833024262
// MI455X (gfx1250) — compile-verified
//
#include <hip/hip_runtime.h>
#include <math.h>

// ---------------- problem constants (match reference) ----------------
#define Nn   100000
#define Ee   800000
#define DIN  256
#define DH   128
#define Kc   16
#define GAMMA 0.1f
#define LAM1  1.0f
#define LAM2  0.5f
#define AZ (1.0f - GAMMA*LAM1 + GAMMA*LAM2)   // 0.95  (z coefficient)
#define CS (GAMMA*LAM1)                       // 0.10  (scatter coefficient)
#define CT (GAMMA*LAM2)                       // 0.05  (Gram coefficient)
#define GRAM_CHUNK 400                        // 250 blocks * 400 rows = 100000
#define GRAM_TILE  16                         // rows of H staged in LDS per iter (8KB)
#define GEMM_KTILE 64                         // K-chunk of A staged in LDS (4KB); divides 128 & 256

typedef __attribute__((ext_vector_type(2))) float v2f;
typedef __attribute__((ext_vector_type(8))) float v8f;

// D(16x16 f32) = A(16x4 f32) x B(4x16 f32) + C   -- native fp32 WMMA on gfx1250
__device__ __forceinline__ v8f wmma4(v2f a, v2f b, v8f c) {
  return __builtin_amdgcn_wmma_f32_16x16x4_f32(
      /*neg_a=*/false, a, /*neg_b=*/false, b,
      /*c_mod=*/(short)0, c, /*reuse_a=*/false, /*reuse_b=*/false);
}

// Fragment layout (ISA 7.12.2, wave32):
//  A[m][k]: lane l holds A[l&15][kb], A[l&15][kb+1]  with kb = 2*(l>>4)
//  B[k][n]: lane l holds B[kb][l&15], B[kb+1][l&15]
//  D[m][n]: vgpr r -> m=r (lanes 0-15) / m=r+8 (lanes 16-31); n = lane&15

// ---------------- utility kernels ----------------
__global__ void fill4_kernel(float4* __restrict__ p, long n4, float v) {
  long i = (long)blockIdx.x * blockDim.x + threadIdx.x;
  long stride = (long)gridDim.x * blockDim.x;
  float4 val = make_float4(v, v, v, v);
  for (; i < n4; i += stride) p[i] = val;
}

// deg initialized to 1.0 (self loop); count targets of the E real edges
__global__ void degree_kernel(const int* __restrict__ edge, float* __restrict__ deg) {
  int e = blockIdx.x * blockDim.x + threadIdx.x;
  if (e < Ee) atomicAdd(&deg[edge[Ee + e]], 1.0f);   // col = edge_index[1]
}

__global__ void rsqrt_kernel(float* __restrict__ deg) {
  int i = blockIdx.x * blockDim.x + threadIdx.x;
  if (i < Nn) {
    float d = deg[i];
    deg[i] = (d > 0.0f) ? rsqrtf(fmaxf(d, 1.0f)) : 0.0f;   // in-place -> dinv
  }
}

// ---------------- C[M x 128] = A[M x Kdim] @ B[Kdim x 128] ----------------
// block = 256 threads = 8 waves; block handles 16 rows; wave w -> cols [16w,16w+16)
// A strip (16 x GEMM_KTILE) staged in LDS (shared by all 8 waves); B read from global (L1-resident)
__global__ void __launch_bounds__(256)
gemm_wmma_kernel(const float* __restrict__ A, const float* __restrict__ B,
                 float* __restrict__ C, int Kdim) {
  __shared__ __align__(16) float atile[16 * GEMM_KTILE];   // 4KB
  int tid  = threadIdx.x;
  int wave = tid >> 5;
  int lane = tid & 31;
  int half = lane >> 4;         // 0/1: selects k-pair and output row half
  int lid  = lane & 15;         // m (A) / n (B,D)
  int kb   = half * 2;
  int row0 = blockIdx.x * 16;
  int col0 = wave * 16;
  int sr   = tid >> 4;          // staging: row 0..15
  int sc   = tid & 15;          // staging: float4 column 0..15
  v8f acc = {};
  for (int k0 = 0; k0 < Kdim; k0 += GEMM_KTILE) {
    __syncthreads();            // previous tile fully consumed
    ((float4*)(atile + sr * GEMM_KTILE))[sc] =
        ((const float4*)(A + (size_t)(row0 + sr) * Kdim + k0))[sc];
    __syncthreads();
#pragma unroll
    for (int ks = 0; ks < GEMM_KTILE; ks += 4) {
      v2f a = *(const v2f*)(atile + lid * GEMM_KTILE + ks + kb);
      v2f b;
      b.x = B[(size_t)(k0 + ks + kb)     * DH + col0 + lid];
      b.y = B[(size_t)(k0 + ks + kb + 1) * DH + col0 + lid];
      acc = wmma4(a, b, acc);
    }
  }
#pragma unroll
  for (int r = 0; r < 8; ++r)
    C[(size_t)(row0 + r + 8 * half) * DH + col0 + lid] = acc[r];
}

// ---------------- sparse propagate: S[c,:] += w * H[r,:] ----------------
// one wave per edge (E real + N self loops); float4 per lane; f32 atomics hit L2 (S is L2-resident)
__global__ void scatter_kernel(const int* __restrict__ edge, const float* __restrict__ dinv,
                               const float* __restrict__ H, float* __restrict__ S) {
  int idx  = blockIdx.x * 8 + (threadIdx.x >> 5);
  int lane = threadIdx.x & 31;
  if (idx >= Ee + Nn) return;
  int r, c; float w;
  if (idx < Ee) {
    r = edge[idx];          // row = source
    c = edge[Ee + idx];     // col = target
    w = dinv[r] * dinv[c];
  } else {
    r = c = idx - Ee;       // self loop
    float di = dinv[r];
    w = di * di;
  }
  const float4 h = *(const float4*)(H + (size_t)r * DH + lane * 4);
  float* sp = S + (size_t)c * DH + lane * 4;
  atomicAdd(sp + 0, w * h.x);
  atomicAdd(sp + 1, w * h.y);
  atomicAdd(sp + 2, w * h.z);
  atomicAdd(sp + 3, w * h.w);
}

// ---------------- G[128x128] += H[chunk]^T @ H[chunk] (split-K atomics) ----------------
// block = 8 waves; wave w -> row tile [16w,16w+16) x all 8 col tiles.
// 16x128 strip of H staged in LDS per iter -> H read exactly once per block (coalesced),
// fragments built from ds loads instead of 8x-redundant global loads.
__global__ void __launch_bounds__(256)
gram_wmma_kernel(const float* __restrict__ H, float* __restrict__ G) {
  __shared__ __align__(16) float tile[GRAM_TILE * DH];   // 8KB
  int tid  = threadIdx.x;
  int wave = tid >> 5;
  int lane = tid & 31;
  int half = lane >> 4;
  int lid  = lane & 15;
  int kb   = 2 * half;
  int n0   = blockIdx.x * GRAM_CHUNK;
  v8f acc[8] = {};
  for (int base = n0; base < n0 + GRAM_CHUNK; base += GRAM_TILE) {
    __syncthreads();            // previous tile fully consumed
    {
      // 16*128 floats = 512 float4; 256 threads -> 2 float4 each (coalesced)
      const float4* src = (const float4*)(H + (size_t)base * DH);
      float4* dst = (float4*)tile;
      dst[tid]       = src[tid];
      dst[tid + 256] = src[tid + 256];
    }
    __syncthreads();
#pragma unroll
    for (int ks = 0; ks < GRAM_TILE; ks += 4) {
      v2f a;                                 // A[m][k] = H[base+ks+k][16w+m]
      a.x = tile[(ks + kb)     * DH + wave * 16 + lid];
      a.y = tile[(ks + kb + 1) * DH + wave * 16 + lid];
#pragma unroll
      for (int ct = 0; ct < 8; ++ct) {
        v2f b;                               // B[k][n] = H[base+ks+k][16ct+n]
        b.x = tile[(ks + kb)     * DH + ct * 16 + lid];
        b.y = tile[(ks + kb + 1) * DH + ct * 16 + lid];
        acc[ct] = wmma4(a, b, acc[ct]);
      }
    }
  }
#pragma unroll
  for (int ct = 0; ct < 8; ++ct)
#pragma unroll
    for (int r = 0; r < 8; ++r)
      atomicAdd(&G[(size_t)(wave * 16 + r + 8 * half) * DH + ct * 16 + lid], acc[ct][r]);
}

// ---------------- fused: S := relu(AZ*H + CS*S - CT*(H@G) + b) (in place over S) ----------------
__global__ void __launch_bounds__(256)
update_wmma_kernel(const float* __restrict__ H, float* __restrict__ S,
                   const float* __restrict__ G, const float* __restrict__ bias) {
  int wave = threadIdx.x >> 5;
  int lane = threadIdx.x & 31;
  int half = lane >> 4;
  int lid  = lane & 15;
  int kb   = 2 * half;
  int row0 = blockIdx.x * 16;
  int col0 = wave * 16;
  v8f acc = {};
  for (int kk = 0; kk < DH; kk += 4) {
    v2f a = *(const v2f*)(H + (size_t)(row0 + lid) * DH + kk + kb);
    v2f b;
    b.x = G[(size_t)(kk + kb)     * DH + col0 + lid];
    b.y = G[(size_t)(kk + kb + 1) * DH + col0 + lid];
    acc = wmma4(a, b, acc);
  }
  float bcol = bias[col0 + lid];
#pragma unroll
  for (int r = 0; r < 8; ++r) {
    size_t off = (size_t)(row0 + r + 8 * half) * DH + col0 + lid;
    float val = AZ * H[off] + CS * S[off] - CT * acc[r] + bcol;
    S[off] = fmaxf(val, 0.0f);
  }
}

// ---------------- logits = X @ Wf + bf ; log_softmax rows ----------------
__global__ void fc_logsoftmax_kernel(const float* __restrict__ X, const float* __restrict__ Wf,
                                     const float* __restrict__ bf, float* __restrict__ out) {
  int n = blockIdx.x * blockDim.x + threadIdx.x;
  if (n >= Nn) return;
  const float* h = X + (size_t)n * DH;
  float logit[Kc];
#pragma unroll
  for (int k = 0; k < Kc; ++k) logit[k] = bf[k];
  for (int j = 0; j < DH; ++j) {
    float hj = h[j];
#pragma unroll
    for (int k = 0; k < Kc; ++k) logit[k] = fmaf(hj, Wf[j * Kc + k], logit[k]);
  }
  float m = logit[0];
#pragma unroll
  for (int k = 1; k < Kc; ++k) m = fmaxf(m, logit[k]);
  float s = 0.0f;
#pragma unroll
  for (int k = 0; k < Kc; ++k) s += expf(logit[k] - m);
  float lse = m + logf(s);
#pragma unroll
  for (int k = 0; k < Kc; ++k) out[(size_t)n * Kc + k] = logit[k] - lse;
}

// ---------------- driver ----------------
extern "C" void kernel_launch(void* const* d_in, const int* in_sizes, int n_in,
                              void* d_out, int out_size, void* d_ws, size_t ws_size,
                              hipStream_t stream) {
  const float* x    = (const float*)d_in[0];
  const int*   edge = (const int*)  d_in[1];
  const float* W1   = (const float*)d_in[2];
  const float* b1   = (const float*)d_in[3];
  const float* W2   = (const float*)d_in[4];
  const float* b2   = (const float*)d_in[5];
  const float* Wf   = (const float*)d_in[6];
  const float* bf   = (const float*)d_in[7];
  float* out = (float*)d_out;

  // workspace layout: H (N*128) | S (N*128, also layer output) | dinv (N) | G (128*128)
  float* H    = (float*)d_ws;
  float* S    = H + (size_t)Nn * DH;
  float* dinv = S + (size_t)Nn * DH;
  float* G    = dinv + Nn;

  dim3 b256(256);

  // gcn_norm: deg = 1 + count(col); dinv = rsqrt(deg)
  fill4_kernel<<<128, b256, 0, stream>>>((float4*)dinv, (long)Nn / 4, 1.0f);
  degree_kernel<<<(Ee + 255) / 256, b256, 0, stream>>>(edge, dinv);
  rsqrt_kernel<<<(Nn + 255) / 256, b256, 0, stream>>>(dinv);

  // ---- layer 1 ----
  gemm_wmma_kernel<<<Nn / 16, b256, 0, stream>>>(x, W1, H, DIN);            // H = x @ W1
  fill4_kernel<<<2048, b256, 0, stream>>>((float4*)S, (long)Nn * DH / 4, 0.0f);
  scatter_kernel<<<(Ee + Nn) / 8, b256, 0, stream>>>(edge, dinv, H, S);     // S = A_norm @ H
  fill4_kernel<<<16, b256, 0, stream>>>((float4*)G, (long)DH * DH / 4, 0.0f);
  gram_wmma_kernel<<<Nn / GRAM_CHUNK, b256, 0, stream>>>(H, G);             // G = H^T H
  update_wmma_kernel<<<Nn / 16, b256, 0, stream>>>(H, S, G, b1);            // S = relu(...)

  // ---- layer 2 (H overwritten, then S reused in place) ----
  gemm_wmma_kernel<<<Nn / 16, b256, 0, stream>>>(S, W2, H, DH);             // H = out1 @ W2
  fill4_kernel<<<2048, b256, 0, stream>>>((float4*)S, (long)Nn * DH / 4, 0.0f);
  scatter_kernel<<<(Ee + Nn) / 8, b256, 0, stream>>>(edge, dinv, H, S);
  fill4_kernel<<<16, b256, 0, stream>>>((float4*)G, (long)DH * DH / 4, 0.0f);
  gram_wmma_kernel<<<Nn / GRAM_CHUNK, b256, 0, stream>>>(H, G);
  update_wmma_kernel<<<Nn / 16, b256, 0, stream>>>(H, S, G, b2);            // S = out2

  // fc + log_softmax
  fc_logsoftmax_kernel<<<(Nn + 255) / 256, b256, 0, stream>>>(S, Wf, bf, out);
}